// Indexer_26637387170050
// MI455X (gfx1250) — compile-verified
//
#include <hip/hip_runtime.h>
#include <hip/hip_bf16.h>

typedef __attribute__((ext_vector_type(2))) float v2f;
typedef __attribute__((ext_vector_type(8))) float v8f;
typedef __attribute__((ext_vector_type(4))) int   v4i;

#define DIM        7168
#define N_HEADS    32
#define HEAD_DIM   128
#define ROPE_DIM   64
#define Q_LORA     1536
#define TOPK       512
#define SEQLEN     2048
#define NUM_SLOTS  4096
#define QDIM       (N_HEADS * HEAD_DIM)   // 4096
#define NEG_INF    (-1.0e9f)

// ---------------------------------------------------------------------------
// gfx1250 async global->LDS staging (ASYNCcnt-tracked), with a synchronous
// fallback if this toolchain does not expose the async builtins.
// Probe result (round 2 stderr): param 0 of the b128 builtin is
//   'int __attribute__((vector_size(16))) __device__ *'  (AS1 int4*),
// so cast to AS-qualified int4 pointers explicitly.
// ---------------------------------------------------------------------------
#if defined(__has_builtin)
# if __has_builtin(__builtin_amdgcn_global_load_async_to_lds_b128) && \
     __has_builtin(__builtin_amdgcn_s_wait_asynccnt)
#  define USE_ASYNC_LDS 1
# endif
#endif

typedef __attribute__((address_space(1))) v4i* gas_v4i_ptr;
typedef __attribute__((address_space(3))) v4i* las_v4i_ptr;

__device__ __forceinline__ void copy_g2l_b128(float* lds_dst, const float* gsrc) {
#if defined(USE_ASYNC_LDS)
    __builtin_amdgcn_global_load_async_to_lds_b128(
        (gas_v4i_ptr)(void*)gsrc,
        (las_v4i_ptr)lds_dst,
        /*offset=*/0, /*cpol=*/0);
#else
    *(float4*)lds_dst = *(const float4*)gsrc;
#endif
}

__device__ __forceinline__ void async_join() {
#if defined(USE_ASYNC_LDS)
    __builtin_amdgcn_s_wait_asynccnt(0);
#endif
}

// ---------------------------------------------------------------------------
// Copy the input cache into the output cache region (slots not overwritten by
// slot_mapping must keep their original values).
// ---------------------------------------------------------------------------
__global__ void copy_cache_kernel(const float* __restrict__ cache,
                                  float* __restrict__ cache_out) {
    int i = blockIdx.x * blockDim.x + threadIdx.x;
    if (i < NUM_SLOTS * HEAD_DIM) cache_out[i] = cache[i];
}

// ---------------------------------------------------------------------------
// k_proj = x @ wk_w  (2048x7168 @ 7168x128), layernorm over 128, interleaved
// RoPE on dims [0,64), write k row to kbuf and scatter into cache_out.
// ---------------------------------------------------------------------------
__global__ __launch_bounds__(128) void k_proj_kernel(
    const float* __restrict__ x, const float* __restrict__ wk,
    const float* __restrict__ cosb, const float* __restrict__ sinb,
    const float* __restrict__ kn_w, const float* __restrict__ kn_b,
    const int* __restrict__ slot_mapping,
    float* __restrict__ kbuf, float* __restrict__ cache_out) {
    const int t = blockIdx.x;
    const int d = threadIdx.x;           // 0..127
    __shared__ float xs[1024];
    __shared__ float red[128];
    __shared__ float red2[128];

    float acc = 0.0f;
    for (int j0 = 0; j0 < DIM; j0 += 1024) {
        for (int u = d; u < 1024; u += 128) xs[u] = x[t * DIM + j0 + u];
        __syncthreads();
        #pragma unroll 8
        for (int u = 0; u < 1024; ++u)
            acc = fmaf(xs[u], wk[(j0 + u) * HEAD_DIM + d], acc);
        __syncthreads();
    }

    red[d]  = acc;
    red2[d] = acc * acc;
    __syncthreads();
    for (int s = 64; s > 0; s >>= 1) {
        if (d < s) { red[d] += red[d + s]; red2[d] += red2[d + s]; }
        __syncthreads();
    }
    const float mu  = red[0] * (1.0f / HEAD_DIM);
    const float var = red2[0] * (1.0f / HEAD_DIM) - mu * mu;
    const float rs  = rsqrtf(var + 1e-6f);
    const float kn  = (acc - mu) * rs * kn_w[d] + kn_b[d];

    __syncthreads();
    red[d] = kn;
    __syncthreads();
    float outv;
    if (d < ROPE_DIM) {
        const float c  = cosb[t * ROPE_DIM + d];
        const float sn = sinb[t * ROPE_DIM + d];
        if ((d & 1) == 0) outv = kn * c - red[d + 1] * sn;   // x1*c - x2*s
        else              outv = kn * c + red[d - 1] * sn;   // x2*c + x1*s
    } else {
        outv = kn;
    }
    kbuf[t * HEAD_DIM + d] = outv;
    cache_out[slot_mapping[t] * HEAD_DIM + d] = outv;
}

// ---------------------------------------------------------------------------
// w = (x @ wproj_w) / 64.  One 256-thread block per token row.
// ---------------------------------------------------------------------------
__global__ __launch_bounds__(256) void wproj_kernel(
    const float* __restrict__ x, const float* __restrict__ wp,
    float* __restrict__ wbuf) {
    const int s   = blockIdx.x;
    const int tid = threadIdx.x;
    const int h     = tid & 31;
    const int chunk = tid >> 5;            // 0..7, each 896 j's
    float acc = 0.0f;
    const int j0 = chunk * (DIM / 8);
    for (int j = j0; j < j0 + DIM / 8; ++j)
        acc = fmaf(x[s * DIM + j], wp[j * N_HEADS + h], acc);
    __shared__ float red[256];
    red[tid] = acc;
    __syncthreads();
    if (tid < 32) {
        float sum = 0.0f;
        #pragma unroll
        for (int c = 0; c < 8; ++c) sum += red[c * 32 + tid];
        wbuf[s * N_HEADS + tid] = sum * 0.015625f;   // 1/(sqrt(32)*sqrt(128))
    }
}

// ---------------------------------------------------------------------------
// q = qr @ wq_b_w  (2048x1536 @ 1536x4096) with V_WMMA_F32_16X16X4_F32.
// 4-wave block computes a 32(s) x 32(c) region; A/B K-panels of depth 64 are
// staged in LDS (async-to-LDS when available), giving 2x reuse per operand
// and ds_load fragment fetches.
//   A frag: lane holds A[sOff + (lane&15)][k4 + 2*(lane>>4) + {0,1}]
//   B frag: lane holds B[k4 + 2*(lane>>4) + {0,1}][cOff + (lane&15)]
//   D frag: VGPR i = D[sOff + i + 8*(lane>>4)][cOff + (lane&15)]
// ---------------------------------------------------------------------------
__global__ __launch_bounds__(128) void q_gemm_kernel(
    const float* __restrict__ qr, const float* __restrict__ wqb,
    float* __restrict__ qbuf) {
    const int tid  = threadIdx.x;
    const int lane = tid & 31;
    const int w    = tid >> 5;
    const int c0 = blockIdx.x * 32;
    const int s0 = blockIdx.y * 32;
    const int rc = lane & 15;
    const int hi = lane >> 4;
    const int kb = hi * 2;
    const int sOff = (w >> 1) * 16;
    const int cOff = (w & 1) * 16;

    __shared__ float At[32 * 64];   // [s-row][k]
    __shared__ float Bt[64 * 32];   // [k][col]

    v8f acc = {};
    for (int kk = 0; kk < Q_LORA; kk += 64) {
        // stage A panel: 32x64 floats = 512 float4 (4 per thread)
        #pragma unroll
        for (int i = 0; i < 4; ++i) {
            const int lin = tid + i * 128;          // 0..511
            const int row = lin >> 4;               // 16 float4 per row
            const int c4  = lin & 15;
            copy_g2l_b128(&At[row * 64 + c4 * 4],
                          &qr[(s0 + row) * Q_LORA + kk + c4 * 4]);
        }
        // stage B panel: 64x32 floats = 512 float4 (4 per thread)
        #pragma unroll
        for (int i = 0; i < 4; ++i) {
            const int lin = tid + i * 128;
            const int row = lin >> 3;               // 8 float4 per row
            const int c4  = lin & 7;
            copy_g2l_b128(&Bt[row * 32 + c4 * 4],
                          &wqb[(kk + row) * QDIM + c0 + c4 * 4]);
        }
        async_join();
        __syncthreads();

        #pragma unroll
        for (int k4 = 0; k4 < 64; k4 += 4) {
            v2f a = *(const v2f*)&At[(sOff + rc) * 64 + k4 + kb];
            v2f b;
            b.x = Bt[(k4 + kb)     * 32 + cOff + rc];
            b.y = Bt[(k4 + kb + 1) * 32 + cOff + rc];
            acc = __builtin_amdgcn_wmma_f32_16x16x4_f32(
                false, a, false, b, (short)0, acc, false, false);
        }
        __syncthreads();
    }
    #pragma unroll
    for (int i = 0; i < 8; ++i)
        qbuf[(s0 + sOff + i + 8 * hi) * QDIM + c0 + cOff + rc] = acc[i];
}

// ---------------------------------------------------------------------------
// Apply interleaved RoPE in-place to q's first 64 dims of each head.
// ---------------------------------------------------------------------------
__global__ void q_rope_kernel(float* __restrict__ qbuf,
                              const float* __restrict__ cosb,
                              const float* __restrict__ sinb) {
    const int idx = blockIdx.x * blockDim.x + threadIdx.x;
    if (idx >= SEQLEN * N_HEADS * (ROPE_DIM / 2)) return;
    const int p = idx & 31;
    const int h = (idx >> 5) & 31;
    const int s = idx >> 10;
    float* qp = qbuf + s * QDIM + h * HEAD_DIM + 2 * p;
    const float x1 = qp[0], x2 = qp[1];
    const float c0 = cosb[s * ROPE_DIM + 2 * p];
    const float c1 = cosb[s * ROPE_DIM + 2 * p + 1];
    const float s0 = sinb[s * ROPE_DIM + 2 * p];
    const float s1 = sinb[s * ROPE_DIM + 2 * p + 1];
    qp[0] = x1 * c0 - x2 * s0;
    qp[1] = x2 * c1 + x1 * s1;
}

// ---------------------------------------------------------------------------
// score[s][t] = sum_h relu(q[s,h,:] . k[t,:]) * w[s,h], causal-masked.
// 4-wave block computes a 32(s) x 32(t) region. The K-tile (32x128) is staged
// in LDS once per block and reused by all 32 heads; the Q-tile (32x128) is
// staged per head. Fully-masked blocks skip all math.
// ---------------------------------------------------------------------------
__global__ __launch_bounds__(128) void score_kernel(
    const float* __restrict__ qbuf, const float* __restrict__ kbuf,
    const float* __restrict__ wbuf, float* __restrict__ sbuf) {
    const int tid  = threadIdx.x;
    const int lane = tid & 31;
    const int w    = tid >> 5;
    const int t0 = blockIdx.x * 32;
    const int s0 = blockIdx.y * 32;
    const int rc = lane & 15;
    const int hi = lane >> 4;
    const int kb = hi * 2;
    const int sOff = (w >> 1) * 16;
    const int tOff = (w & 1) * 16;

    if (t0 > s0 + 31) {                     // fully masked block
        #pragma unroll
        for (int i = 0; i < 8; ++i)
            sbuf[(s0 + sOff + i + 8 * hi) * SEQLEN + t0 + tOff + rc] = NEG_INF;
        return;
    }

    __shared__ float ktile[32 * HEAD_DIM];  // 16 KB
    __shared__ float qtile[32 * HEAD_DIM];  // 16 KB

    // stage K tile once: 32x128 floats = 1024 float4 (8 per thread)
    #pragma unroll
    for (int i = 0; i < 8; ++i) {
        const int lin = tid + i * 128;      // 0..1023
        const int row = lin >> 5;           // 32 float4 per row
        const int c4  = lin & 31;
        copy_g2l_b128(&ktile[row * HEAD_DIM + c4 * 4],
                      &kbuf[(t0 + row) * HEAD_DIM + c4 * 4]);
    }

    v8f score = {};
    for (int h = 0; h < N_HEADS; ++h) {
        // stage Q tile for this head
        #pragma unroll
        for (int i = 0; i < 8; ++i) {
            const int lin = tid + i * 128;
            const int row = lin >> 5;
            const int c4  = lin & 31;
            copy_g2l_b128(&qtile[row * HEAD_DIM + c4 * 4],
                          &qbuf[(s0 + row) * QDIM + h * HEAD_DIM + c4 * 4]);
        }
        async_join();
        __syncthreads();

        v8f lg = {};
        #pragma unroll
        for (int kk = 0; kk < HEAD_DIM; kk += 4) {
            v2f a = *(const v2f*)&qtile[(sOff + rc) * HEAD_DIM + kk + kb];
            v2f b = *(const v2f*)&ktile[(tOff + rc) * HEAD_DIM + kk + kb];
            lg = __builtin_amdgcn_wmma_f32_16x16x4_f32(
                false, a, false, b, (short)0, lg, false, false);
        }
        #pragma unroll
        for (int i = 0; i < 8; ++i) {
            const float wv = wbuf[(s0 + sOff + i + 8 * hi) * N_HEADS + h];
            score[i] += fmaxf(lg[i], 0.0f) * wv;
        }
        __syncthreads();                    // before qtile is overwritten
    }
    #pragma unroll
    for (int i = 0; i < 8; ++i) {
        const int srow = s0 + sOff + i + 8 * hi;
        const int tcol = t0 + tOff + rc;
        sbuf[srow * SEQLEN + tcol] = (tcol <= srow) ? score[i] : NEG_INF;
    }
}

// ---------------------------------------------------------------------------
// Exact sorted top-512 per row via bitonic sort of 2048 packed keys in LDS.
// key = orderable(f32) << 32 | (~idx)  -> descending sort gives jax.top_k
// semantics (ties broken by lowest index).
// ---------------------------------------------------------------------------
__global__ __launch_bounds__(1024) void topk_kernel(
    const float* __restrict__ sbuf,
    float* __restrict__ out_idx, float* __restrict__ out_val) {
    const int s   = blockIdx.x;
    const int tid = threadIdx.x;
    __shared__ unsigned long long kv[SEQLEN];

    for (int e = tid; e < SEQLEN; e += 1024) {
        const float v = sbuf[s * SEQLEN + e];
        unsigned u = __float_as_uint(v);
        u = (u & 0x80000000u) ? ~u : (u | 0x80000000u);   // order-preserving
        kv[e] = ((unsigned long long)u << 32) |
                (unsigned long long)(0xFFFFFFFFu - (unsigned)e);
    }
    __syncthreads();

    for (int k = 2; k <= SEQLEN; k <<= 1) {
        for (int j = k >> 1; j > 0; j >>= 1) {
            for (int i = tid; i < SEQLEN; i += 1024) {
                const int p = i ^ j;
                if (p > i) {
                    const bool desc = ((i & k) == 0);
                    const unsigned long long a = kv[i], b = kv[p];
                    if (desc ? (a < b) : (a > b)) { kv[i] = b; kv[p] = a; }
                }
            }
            __syncthreads();
        }
    }

    if (tid < TOPK) {
        const unsigned long long key = kv[tid];
        unsigned u = (unsigned)(key >> 32);
        u = (u & 0x80000000u) ? (u ^ 0x80000000u) : ~u;   // un-map
        const unsigned idx = 0xFFFFFFFFu - (unsigned)(key & 0xFFFFFFFFu);
        out_idx[s * TOPK + tid] = (float)idx;             // d_out is float
        out_val[s * TOPK + tid] = __uint_as_float(u);
    }
}

// ---------------------------------------------------------------------------
extern "C" void kernel_launch(void* const* d_in, const int* in_sizes, int n_in,
                              void* d_out, int out_size, void* d_ws, size_t ws_size,
                              hipStream_t stream) {
    const float* x    = (const float*)d_in[0];
    const float* qr   = (const float*)d_in[1];
    const float* cosb = (const float*)d_in[2];
    const float* sinb = (const float*)d_in[3];
    const float* cache = (const float*)d_in[4];
    const int*   slot  = (const int*)d_in[5];
    const float* wqb  = (const float*)d_in[6];
    const float* wk   = (const float*)d_in[7];
    const float* wp   = (const float*)d_in[8];
    const float* kn_w = (const float*)d_in[9];
    const float* kn_b = (const float*)d_in[10];

    // workspace layout (floats): k | q | w | score  == ~49.3 MB
    float* kbuf = (float*)d_ws;
    float* qbuf = kbuf + SEQLEN * HEAD_DIM;
    float* wbuf = qbuf + (size_t)SEQLEN * QDIM;
    float* sbuf = wbuf + SEQLEN * N_HEADS;

    float* out      = (float*)d_out;
    float* out_idx  = out;                                 // topk_idx
    float* out_val  = out + (size_t)SEQLEN * TOPK;         // topk_vals
    float* cacheout = out + (size_t)2 * SEQLEN * TOPK;     // cache_new

    copy_cache_kernel<<<(NUM_SLOTS * HEAD_DIM + 255) / 256, 256, 0, stream>>>(
        cache, cacheout);

    k_proj_kernel<<<SEQLEN, 128, 0, stream>>>(
        x, wk, cosb, sinb, kn_w, kn_b, slot, kbuf, cacheout);

    wproj_kernel<<<SEQLEN, 256, 0, stream>>>(x, wp, wbuf);

    q_gemm_kernel<<<dim3(QDIM / 32, SEQLEN / 32), 128, 0, stream>>>(
        qr, wqb, qbuf);

    const int nrope = SEQLEN * N_HEADS * (ROPE_DIM / 2);
    q_rope_kernel<<<(nrope + 255) / 256, 256, 0, stream>>>(qbuf, cosb, sinb);

    score_kernel<<<dim3(SEQLEN / 32, SEQLEN / 32), 128, 0, stream>>>(
        qbuf, kbuf, wbuf, sbuf);

    topk_kernel<<<SEQLEN, 1024, 0, stream>>>(sbuf, out_idx, out_val);
}